// NeRFRendererDGS_67181878444949
// MI455X (gfx1250) — compile-verified
//
#include <hip/hip_runtime.h>
#include <math.h>

#define SBN 2
#define NVN 3
#define NRN 2048
#define NCN 1000
#define NSN 40
#define NGN 15
#define NKEEP (NSN - NGN)   // 25
#define HGT 256
#define WID 256
#define HID 256
#define HID_P 257           // padded LDS stride (bank-conflict free)
#define DDM 0.05f

typedef __attribute__((ext_vector_type(2))) float v2f;
typedef __attribute__((ext_vector_type(8))) float v8f;

__device__ __forceinline__ float zstep(int c, float u) {
    // jnp.linspace(0, 1-1/NC, NC)[c] + u/NC
    const float lin  = (1.0f - 1.0f / (float)NCN) / (float)(NCN - 1);
    const float step = 1.0f / (float)NCN;
    return (float)c * lin + u * step;
}

// ---------------------------------------------------------------------------
// Kernel A: per-candidate likelihood, max over views  -> pt[SB*NR*NC]
// ---------------------------------------------------------------------------
__global__ void __launch_bounds__(256) lik_kernel(
    const float* __restrict__ rays, const float* __restrict__ poses,
    const float* __restrict__ focal, const float* __restrict__ cc,
    const float* __restrict__ depth_map, const float* __restrict__ std_map,
    const float* __restrict__ normal_map, const float* __restrict__ u_coarse,
    float* __restrict__ pt_out)
{
    long idx = (long)blockIdx.x * blockDim.x + threadIdx.x;
    const long total = (long)SBN * NRN * NCN;
    if (idx >= total) return;
    int  c  = (int)(idx % NCN);
    long rr = idx / NCN;              // sb*NR + ray
    int  sb = (int)(rr / NRN);

    const float* rp = rays + rr * 8;
    float ox = rp[0], oy = rp[1], oz = rp[2];
    float dx = rp[3], dy = rp[4], dz = rp[5];
    float nearv = rp[6], farv = rp[7];

    float u  = u_coarse[rr * NCN + c];
    float zs = zstep(c, u);
    float z  = nearv * (1.0f - zs) + farv * zs;
    float wx = ox + z * dx, wy = oy + z * dy, wz = oz + z * dz;
    float ss = (farv - nearv) / (float)NCN;

    float pt = 0.0f;
    for (int v = 0; v < NVN; ++v) {
        const float* P = poses + (long)(sb * NVN + v) * 16;
        float cx = P[0] * wx + P[1] * wy + P[2]  * wz + P[3];
        float cy = P[4] * wx + P[5] * wy + P[6]  * wz + P[7];
        float cz = P[8] * wx + P[9] * wy + P[10] * wz + P[11];
        float rdx = P[0] * dx + P[1] * dy + P[2]  * dz;
        float rdy = P[4] * dx + P[5] * dy + P[6]  * dz;
        float rdz = P[8] * dx + P[9] * dy + P[10] * dz;
        float fx  = focal[(sb * NVN + v) * 2 + 0], fy = focal[(sb * NVN + v) * 2 + 1];
        float cxo = cc[(sb * NVN + v) * 2 + 0],    cyo = cc[(sb * NVN + v) * 2 + 1];

        float iu = cx / cz, iv = cy / cz;
        float un = (iu * fx + cxo) / (float)WID * 2.0f - 1.0f;
        float vn = (iv * fy + cyo) / (float)HGT * 2.0f - 1.0f;
        // jnp.round is round-half-even -> rintf
        float xf = rintf((un + 1.0f) * 0.5f * (float)WID - 0.5f);
        float yf = rintf((vn + 1.0f) * 0.5f * (float)HGT - 0.5f);
        int xi = (int)fminf(fmaxf(xf, 0.0f), (float)(WID - 1));
        int yi = (int)fminf(fmaxf(yf, 0.0f), (float)(HGT - 1));
        long lin = (long)yi * WID + xi;

        long mb = (long)(sb * NVN + v) * (HGT * WID);
        float rd = depth_map[mb + lin];
        float rs = std_map[mb + lin];
        long nb = (long)(sb * NVN + v) * 3 * (HGT * WID);
        float n0 = normal_map[nb + lin];
        float n1 = normal_map[nb + (long)HGT * WID + lin];
        float n2 = normal_map[nb + 2L * HGT * WID + lin];

        float cosd = rdx * n0 + rdy * n1 + rdz * n2;
        bool  mask = (rs != 0.0f) && (fabsf(rd - cz) < DDM) && (cosd <= 0.0f);
        float inv  = 1.0f / ((mask ? rs : 1.0f) * 1.41421356237309504880f);
        float e1 = erff((cz + 0.5f * ss - rd) * inv);
        float e2 = erff((cz - 0.5f * ss - rd) * inv);
        float lik = 0.5f * fabsf(e1 - e2);
        pt = fmaxf(pt, mask ? lik : 0.0f);
    }
    pt_out[idx] = pt;
}

// ---------------------------------------------------------------------------
// Kernel B: per-ray top-k selection, gaussian augmentation, fill, sort -> z[40]
// ---------------------------------------------------------------------------
__global__ void __launch_bounds__(64) sample_kernel(
    const float* __restrict__ rays, const float* __restrict__ u_coarse,
    const float* __restrict__ g_noise, const float* __restrict__ u_fill,
    const float* __restrict__ pt_buf, float* __restrict__ z_out)
{
    int rr = blockIdx.x * blockDim.x + threadIdx.x;
    if (rr >= SBN * NRN) return;
    const float* rp = rays + (long)rr * 8;
    float nearv = rp[6], farv = rp[7];
    const float* ptp = pt_buf + (long)rr * NCN;
    const float* up  = u_coarse + (long)rr * NCN;

    // stable top-25 (desc value, tie -> lower index) == first 25 of lax.top_k
    float tv[NKEEP]; int ti[NKEEP];
    int cnt = 0;
    float T = 1.0f, wsum = 0.0f, wz = 0.0f;
    bool anym = false;
    for (int c = 0; c < NCN; ++c) {
        float p  = ptp[c];
        float zs = zstep(c, up[c]);
        float z  = nearv * (1.0f - zs) + farv * zs;
        float op = p * T;
        T *= (1.0f - p);
        wsum += op; wz += z * op;
        anym = anym || (op != 0.0f);
        if (cnt == NKEEP && p <= tv[NKEEP - 1]) continue;
        int pos = 0;
        while (pos < cnt && tv[pos] >= p) ++pos;
        if (pos < NKEEP) {
            int end = (cnt < NKEEP) ? cnt : (NKEEP - 1);
            for (int j = end; j > pos; --j) { tv[j] = tv[j - 1]; ti[j] = ti[j - 1]; }
            tv[pos] = p; ti[pos] = c;
            if (cnt < NKEEP) ++cnt;
        }
    }
    float sw   = (wsum > 0.0f) ? wsum : 1.0f;
    float mean = wz / sw;
    // second pass: variance around mean (matches reference formulation)
    T = 1.0f; float var = 0.0f;
    for (int c = 0; c < NCN; ++c) {
        float p  = ptp[c];
        float zs = zstep(c, up[c]);
        float z  = nearv * (1.0f - zs) + farv * zs;
        float op = p * T; T *= (1.0f - p);
        float d = z - mean;
        var += op * d * d;
    }
    float stdv = sqrtf(var / sw);

    float zsel[NSN];
    for (int j = 0; j < NKEEP; ++j) {
        if (tv[j] == 0.0f) zsel[j] = 0.0f;
        else {
            int c = ti[j];
            float zs = zstep(c, up[c]);
            zsel[j] = nearv * (1.0f - zs) + farv * zs;
        }
    }
    for (int j = 0; j < NGN; ++j) {
        float g = g_noise[(long)rr * NGN + j];
        zsel[NKEEP + j] = anym ? (g * stdv + mean) : 0.0f;
    }
    // sort ascending
    for (int i = 1; i < NSN; ++i) {
        float key = zsel[i]; int j = i - 1;
        while (j >= 0 && zsel[j] > key) { zsel[j + 1] = zsel[j]; --j; }
        zsel[j + 1] = key;
    }
    int nmiss = 0;
    for (int i = 0; i < NSN; ++i) if (zsel[i] == 0.0f) ++nmiss;
    float stepf = (farv - nearv) / (float)((nmiss > 1) ? nmiss : 1);
    for (int i = 0; i < NSN; ++i) {
        if (zsel[i] == 0.0f) {
            float uf = u_fill[(long)rr * NSN + i];
            zsel[i] = nearv + (float)i * stepf + uf * stepf;
        }
    }
    for (int i = 1; i < NSN; ++i) {
        float key = zsel[i]; int j = i - 1;
        while (j >= 0 && zsel[j] > key) { zsel[j + 1] = zsel[j]; --j; }
        zsel[j + 1] = key;
    }
    for (int i = 0; i < NSN; ++i) z_out[(long)rr * NSN + i] = zsel[i];
}

// ---------------------------------------------------------------------------
// Kernel C: MLP via V_WMMA_F32_16X16X4_F32. One wave = 16 points (M),
// K = 6 features (2 K-blocks of 4, zero-padded), N = 256 hidden (16 tiles).
// Branch-free zero padding: loads are always in-bounds, masked by multiply.
// ---------------------------------------------------------------------------
__global__ void __launch_bounds__(64) mlp_kernel(
    const float* __restrict__ rays, const float* __restrict__ z_buf,
    const float* __restrict__ W1, const float* __restrict__ B1,
    const float* __restrict__ W2, const float* __restrict__ B2,
    float* __restrict__ out_buf)
{
    __shared__ float hbuf[2][16 * HID_P];
    int lane = threadIdx.x & 31;
    int wave = threadIdx.x >> 5;
    int tile = blockIdx.x * 2 + wave;
    int m    = lane & 15;
    int half = lane >> 4;
    float hs = half ? 0.0f : 1.0f;      // mask for K=6,7 zero padding

    long p  = (long)tile * 16 + m;      // point handled by this lane slot
    long rr = p / NSN;
    const float* rp = rays + rr * 8;
    float z = z_buf[p];
    float f0 = rp[0] + z * rp[3];
    float f1 = rp[1] + z * rp[4];
    float f2 = rp[2] + z * rp[5];
    float f3 = rp[3], f4 = rp[4], f5 = rp[5];

    // A layout (16x4 f32): lanes 0-15 hold K=0,1 ; lanes 16-31 hold K=2,3
    v2f a0, a1;
    a0.x = half ? f2 : f0;  a0.y = half ? f3 : f1;   // K-block 0..3
    a1.x = f4 * hs;         a1.y = f5 * hs;          // K-block 4..7 (6,7 = pad)

    int brow = half * 2;                // B rows for this lane half
    const float* W1r0 = W1 + (brow + 0) * HID;
    const float* W1r1 = W1 + (brow + 1) * HID;
    const float* W1r4 = W1 + 4 * HID;
    const float* W1r5 = W1 + 5 * HID;

    for (int t = 0; t < 16; ++t) {
        int col = t * 16 + m;           // N column
        // B layout (4x16 f32): VGPR0 = rows {0,2}, VGPR1 = rows {1,3} per half
        v2f b0, b1v;
        b0.x  = W1r0[col];
        b0.y  = W1r1[col];
        b1v.x = W1r4[col] * hs;         // rows 6,7 are zero padding
        b1v.y = W1r5[col] * hs;

        v8f acc = {0.f, 0.f, 0.f, 0.f, 0.f, 0.f, 0.f, 0.f};
        acc = __builtin_amdgcn_wmma_f32_16x16x4_f32(false, a0, false, b0,
                                                    (short)0, acc, false, false);
        acc = __builtin_amdgcn_wmma_f32_16x16x4_f32(false, a1, false, b1v,
                                                    (short)0, acc, false, false);
        float bias = B1[col];
#pragma unroll
        for (int r = 0; r < 8; ++r) {
            float h = fmaxf(acc[r] + bias, 0.0f);    // +b1, ReLU
            int mrow = r + half * 8;                  // C layout M index
            hbuf[wave][mrow * HID_P + col] = h;
        }
    }
    __syncthreads();

    // 256 -> 4 dense head: lanes 0-15 each own one point, full f32 dot
    if (half == 0) {
        float acc0 = B2[0], acc1 = B2[1], acc2 = B2[2], acc3 = B2[3];
        for (int n = 0; n < HID; ++n) {
            float h = hbuf[wave][m * HID_P + n];
            acc0 += h * W2[n * 4 + 0];
            acc1 += h * W2[n * 4 + 1];
            acc2 += h * W2[n * 4 + 2];
            acc3 += h * W2[n * 4 + 3];
        }
        long pp = (long)tile * 16 + m;
        out_buf[pp * 4 + 0] = acc0; out_buf[pp * 4 + 1] = acc1;
        out_buf[pp * 4 + 2] = acc2; out_buf[pp * 4 + 3] = acc3;
    }
}

// ---------------------------------------------------------------------------
// Kernel D: alpha compositing, one thread per ray
// ---------------------------------------------------------------------------
__global__ void __launch_bounds__(64) comp_kernel(
    const float* __restrict__ rays, const float* __restrict__ z_buf,
    const float* __restrict__ mlp, float* __restrict__ outp)
{
    int rr = blockIdx.x * blockDim.x + threadIdx.x;
    if (rr >= SBN * NRN) return;
    float farv = rays[(long)rr * 8 + 7];
    float z[NSN];
    for (int i = 0; i < NSN; ++i) z[i] = z_buf[(long)rr * NSN + i];
    float T = 1.0f, wtot = 0.0f, r0 = 0.f, r1 = 0.f, r2 = 0.f, dep = 0.f;
    for (int i = 0; i < NSN; ++i) {
        float delta = (i < NSN - 1) ? (z[i + 1] - z[i]) : (farv - z[NSN - 1]);
        const float* o = mlp + ((long)rr * NSN + i) * 4;
        float a = 1.0f - expf(-delta * fmaxf(o[3], 0.0f));
        float w = a * T;
        outp[(long)rr * NSN + i] = w;
        r0 += w * o[0]; r1 += w * o[1]; r2 += w * o[2];
        dep += w * z[i];
        wtot += w;
        T *= (1.0f - a + 1e-10f);
    }
    float add = 1.0f - wtot;
    float* rgbp = outp + (long)SBN * NRN * NSN;
    rgbp[(long)rr * 3 + 0] = r0 + add;
    rgbp[(long)rr * 3 + 1] = r1 + add;
    rgbp[(long)rr * 3 + 2] = r2 + add;
    float* depp = rgbp + (long)SBN * NRN * 3;
    depp[rr] = dep;
}

// ---------------------------------------------------------------------------
extern "C" void kernel_launch(void* const* d_in, const int* in_sizes, int n_in,
                              void* d_out, int out_size, void* d_ws, size_t ws_size,
                              hipStream_t stream) {
    (void)in_sizes; (void)n_in; (void)out_size; (void)ws_size;
    const float* rays       = (const float*)d_in[0];
    const float* poses      = (const float*)d_in[1];
    const float* focal      = (const float*)d_in[2];
    const float* cc         = (const float*)d_in[3];
    const float* depth_map  = (const float*)d_in[4];
    const float* std_map    = (const float*)d_in[5];
    const float* normal_map = (const float*)d_in[6];
    const float* W1         = (const float*)d_in[7];
    const float* B1         = (const float*)d_in[8];
    const float* W2         = (const float*)d_in[9];
    const float* B2         = (const float*)d_in[10];
    const float* u_coarse   = (const float*)d_in[11];
    const float* g_noise    = (const float*)d_in[12];
    const float* u_fill     = (const float*)d_in[13];

    float* ws      = (float*)d_ws;
    float* pt_buf  = ws;                                      // SB*NR*NC
    float* z_buf   = pt_buf + (size_t)SBN * NRN * NCN;        // SB*NR*NS
    float* mlp_buf = z_buf + (size_t)SBN * NRN * NSN;         // SB*NR*NS*4

    {
        long total = (long)SBN * NRN * NCN;                   // 4,096,000
        int blk = 256;
        lik_kernel<<<dim3((unsigned)((total + blk - 1) / blk)), dim3(blk), 0, stream>>>(
            rays, poses, focal, cc, depth_map, std_map, normal_map, u_coarse, pt_buf);
    }
    {
        int total = SBN * NRN, blk = 64;
        sample_kernel<<<dim3((total + blk - 1) / blk), dim3(blk), 0, stream>>>(
            rays, u_coarse, g_noise, u_fill, pt_buf, z_buf);
    }
    {
        int ntiles = SBN * NRN * NSN / 16;                    // 10240, 2 waves/block
        mlp_kernel<<<dim3(ntiles / 2), dim3(64), 0, stream>>>(
            rays, z_buf, W1, B1, W2, B2, mlp_buf);
    }
    {
        int total = SBN * NRN, blk = 64;
        comp_kernel<<<dim3((total + blk - 1) / blk), dim3(blk), 0, stream>>>(
            rays, z_buf, mlp_buf, (float*)d_out);
    }
}